// DeformableConv_59270548685491
// MI455X (gfx1250) — compile-verified
//
#include <hip/hip_runtime.h>
#include <hip/hip_bf16.h>
#include <math.h>

// ---------------------------------------------------------------------------
// Deformable conv (DCNv1), fused for gfx1250.
//   B=8, Cin=64, H=W=128, Cout=64, K=3, stride=1, pad=1, dil=1
// Main GEMM: out[co,pix] = sum_{k'} A[co,k'] * vals[k',pix],  K=576, N=131072
// K order permuted to k' = kk*64 + ci so one WMMA K-step (4 k') shares a tap.
// ---------------------------------------------------------------------------

typedef __attribute__((ext_vector_type(2))) float v2f;
typedef __attribute__((ext_vector_type(8))) float v8f;

#define CIN   64
#define COUT  64
#define HH    128
#define WW    128
#define HW    (HH * WW)          // 16384
#define NOFF  18                 // 2*K*K offset channels
#define KDIM  576                // Cin * 9
#define KB_N  144                // KDIM / 4  (WMMA K-steps)

// ---------------------------------------------------------------------------
// Kernel 1: offset conv.  One thread per output pixel, all 18 channels.
// w_off (18*64*9 floats = 41.5 KB) staged in LDS; 3x3 input patch kept in
// registers and reused across the 18 output channels.
// ---------------------------------------------------------------------------
__global__ __launch_bounds__(256) void offconv_kernel(
    const float* __restrict__ x, const float* __restrict__ w_off,
    const float* __restrict__ b_off, float* __restrict__ off_out) {
  __shared__ float wl[NOFF * CIN * 9];
  for (int i = threadIdx.x; i < NOFF * CIN * 9; i += 256) wl[i] = w_off[i];
  __syncthreads();

  const int p  = blockIdx.x * 256 + threadIdx.x;   // 0 .. 131071
  const int wo = p & 127;
  const int ho = (p >> 7) & 127;
  const int b  = p >> 14;

  float acc[NOFF];
#pragma unroll
  for (int c = 0; c < NOFF; ++c) acc[c] = b_off[c];

  const float* xb = x + (size_t)b * CIN * HW;
#pragma unroll 1
  for (int ci = 0; ci < CIN; ++ci) {
    const float* xc = xb + ci * HW;
    float patch[9];
#pragma unroll
    for (int r = 0; r < 3; ++r) {
      const int h = ho - 1 + r;
      const bool hv = (unsigned)h < (unsigned)HH;
#pragma unroll
      for (int q = 0; q < 3; ++q) {
        const int w = wo - 1 + q;
        const bool wv = (unsigned)w < (unsigned)WW;
        patch[r * 3 + q] = (hv && wv) ? xc[h * WW + w] : 0.0f;
      }
    }
#pragma unroll
    for (int c = 0; c < NOFF; ++c) {
      const float* ww = &wl[(c * CIN + ci) * 9];
      float s = 0.0f;
#pragma unroll
      for (int kk = 0; kk < 9; ++kk) s = fmaf(patch[kk], ww[kk], s);
      acc[c] += s;
    }
  }
#pragma unroll
  for (int c = 0; c < NOFF; ++c)
    off_out[(((size_t)b * NOFF + c) << 14) + (ho << 7) + wo] = acc[c];
}

// ---------------------------------------------------------------------------
// Kernel 2: pack w_def into WMMA A-fragment order.
// A (16x4 f32) ISA layout: lane l holds M = l%16; VGPR j holds K = 2*(l/16)+j.
// apack[cb][kb][lane][j],  co = cb*16 + l%16,  k' = kb*4 + 2*(l/16) + j,
// ci = k' % 64, kk = k' / 64  (permuted K order).
// ---------------------------------------------------------------------------
__global__ __launch_bounds__(256) void pack_wdef_kernel(
    const float* __restrict__ w_def, float* __restrict__ apack) {
  const int i = blockIdx.x * 256 + threadIdx.x;    // 4*144*64 = 36864 total
  if (i >= 4 * KB_N * 64) return;
  const int j    = i & 1;
  const int lane = (i >> 1) & 31;
  const int rest = i >> 6;
  const int kb   = rest % KB_N;
  const int cb   = rest / KB_N;
  const int co   = cb * 16 + (lane & 15);
  const int kp   = kb * 4 + 2 * (lane >> 4) + j;
  const int ci   = kp & 63;
  const int kk   = kp >> 6;
  apack[i] = w_def[(co * CIN + ci) * 9 + kk];
}

// ---------------------------------------------------------------------------
// Kernel 3: fused bilinear sampling + WMMA GEMM.
// One wave -> 16 pixels x 64 Cout (4 accumulator chains of v_wmma_f32_16x16x4).
// Tap kk is uniform across a whole group of 16 K-steps, so bilinear coords /
// weights are computed once per tap and reused for all 64 input channels.
// B fragment built in registers: VGPR j = row k_local = 2*(lane/16)+j,
// col = lane%16 (mirrors the A K-layout).
// ---------------------------------------------------------------------------
__global__ __launch_bounds__(256) void deform_wmma_kernel(
    const float* __restrict__ x, const float* __restrict__ off,
    const float* __restrict__ apack, const float* __restrict__ b_def,
    float* __restrict__ out) {
  const int lane = threadIdx.x & 31;
  const int wave = threadIdx.x >> 5;
  const int grp  = lane >> 4;          // 0 or 1
  const int px   = lane & 15;

  const int p  = (blockIdx.x * 8 + wave) * 16 + px;   // pixel for this lane col
  const int wo = p & 127;
  const int ho = (p >> 7) & 127;
  const int b  = p >> 14;

  const float* xb = x + (size_t)b * CIN * HW;
  const int pix_off = (ho << 7) + wo;

  v8f acc0 = {}, acc1 = {}, acc2 = {}, acc3 = {};

#pragma unroll 1
  for (int kk = 0; kk < 9; ++kk) {
    // --- per-tap bilinear setup (reused for 16 WMMA steps / 64 channels) ---
    const float dy = off[(((size_t)b * NOFF + 2 * kk) << 14) + pix_off];
    const float dx = off[(((size_t)b * NOFF + 2 * kk + 1) << 14) + pix_off];
    const float py  = (float)(ho - 1 + kk / 3) + dy;
    const float pxf = (float)(wo - 1 + kk % 3) + dx;
    const float h0f = floorf(py), w0f = floorf(pxf);
    const float lh = py - h0f, lw = pxf - w0f;
    const int h0 = (int)h0f, w0 = (int)w0f;
    const int h1 = h0 + 1,   w1 = w0 + 1;
    const bool h0v = (unsigned)h0 < (unsigned)HH;
    const bool h1v = (unsigned)h1 < (unsigned)HH;
    const bool w0v = (unsigned)w0 < (unsigned)WW;
    const bool w1v = (unsigned)w1 < (unsigned)WW;
    // fold validity into weights: gather at clamped addr, weight=0 if OOB
    const float g00 = (h0v && w0v) ? (1.0f - lh) * (1.0f - lw) : 0.0f;
    const float g01 = (h0v && w1v) ? (1.0f - lh) * lw          : 0.0f;
    const float g10 = (h1v && w0v) ? lh * (1.0f - lw)          : 0.0f;
    const float g11 = (h1v && w1v) ? lh * lw                   : 0.0f;
    const int h0c = h0 < 0 ? 0 : (h0 > HH - 1 ? HH - 1 : h0);
    const int h1c = h1 < 0 ? 0 : (h1 > HH - 1 ? HH - 1 : h1);
    const int w0c = w0 < 0 ? 0 : (w0 > WW - 1 ? WW - 1 : w0);
    const int w1c = w1 < 0 ? 0 : (w1 > WW - 1 ? WW - 1 : w1);
    const int o00 = h0c * WW + w0c, o01 = h0c * WW + w1c;
    const int o10 = h1c * WW + w0c, o11 = h1c * WW + w1c;

#pragma unroll 4
    for (int s = 0; s < 16; ++s) {               // kb = kk*16 + s
      const int kb  = kk * 16 + s;
      const int ci0 = s * 4 + 2 * grp;           // ci = k' % 64
      v2f bfrag;
#pragma unroll
      for (int j = 0; j < 2; ++j) {
        const float* xc = xb + (ci0 + j) * HW;
        float v = g00 * xc[o00];
        v = fmaf(g01, xc[o01], v);
        v = fmaf(g10, xc[o10], v);
        v = fmaf(g11, xc[o11], v);
        bfrag[j] = v;
      }
      const float* ap = apack + (size_t)kb * 64 + lane * 2;
      const v2f a0 = *(const v2f*)(ap);
      const v2f a1 = *(const v2f*)(ap + 1 * KB_N * 64);
      const v2f a2 = *(const v2f*)(ap + 2 * KB_N * 64);
      const v2f a3 = *(const v2f*)(ap + 3 * KB_N * 64);
      acc0 = __builtin_amdgcn_wmma_f32_16x16x4_f32(false, a0, false, bfrag,
                                                   (short)0, acc0, false, false);
      acc1 = __builtin_amdgcn_wmma_f32_16x16x4_f32(false, a1, false, bfrag,
                                                   (short)0, acc1, false, false);
      acc2 = __builtin_amdgcn_wmma_f32_16x16x4_f32(false, a2, false, bfrag,
                                                   (short)0, acc2, false, false);
      acc3 = __builtin_amdgcn_wmma_f32_16x16x4_f32(false, a3, false, bfrag,
                                                   (short)0, acc3, false, false);
    }
  }

  // C/D layout: VGPR r, lanes 0-15 -> M=r, lanes 16-31 -> M=r+8; N = lane%16.
#pragma unroll
  for (int cb = 0; cb < 4; ++cb) {
    const v8f a = (cb == 0) ? acc0 : (cb == 1) ? acc1 : (cb == 2) ? acc2 : acc3;
#pragma unroll
    for (int r = 0; r < 8; ++r) {
      const int co = cb * 16 + r + 8 * grp;
      out[(((size_t)b * COUT + co) << 14) + pix_off] = a[r] + b_def[co];
    }
  }
}

// ---------------------------------------------------------------------------
extern "C" void kernel_launch(void* const* d_in, const int* in_sizes, int n_in,
                              void* d_out, int out_size, void* d_ws, size_t ws_size,
                              hipStream_t stream) {
  const float* x     = (const float*)d_in[0];
  const float* w_off = (const float*)d_in[1];
  const float* b_off = (const float*)d_in[2];
  const float* w_def = (const float*)d_in[3];
  const float* b_def = (const float*)d_in[4];
  float* out = (float*)d_out;

  float* off_ws = (float*)d_ws;                       // 8*18*16384 floats (9.4 MB)
  float* apack  = off_ws + (size_t)8 * NOFF * HW;     // 4*144*64 floats (147 KB)

  offconv_kernel<<<512, 256, 0, stream>>>(x, w_off, b_off, off_ws);
  pack_wdef_kernel<<<144, 256, 0, stream>>>(w_def, apack);
  deform_wmma_kernel<<<1024, 256, 0, stream>>>(x, off_ws, apack, b_def, out);
}